// ArmatureLayer_2920577761894
// MI455X (gfx1250) — compile-verified
//
#include <hip/hip_runtime.h>
#include <hip/hip_bf16.h>

#define NUM_BONES 64

typedef __attribute__((ext_vector_type(2))) float v2f;
typedef __attribute__((ext_vector_type(4))) float v4f;
typedef __attribute__((ext_vector_type(8))) float v8f;

// ---------------------------------------------------------------------------
// Kernel 1: sequential bone-chain -> T[64][4][4]; also emits the B-operand
// table pre-swizzled into the V_WMMA_F32_16X16X4_F32 B-matrix register layout.
// Btab layout: Btab[b*64 + 2*L + 0] = B-VGPR0 value for lane L
//              Btab[b*64 + 2*L + 1] = B-VGPR1 value for lane L
// where B(K,N) = T[b][K][N] (N<4), zero for N=4..15;
// lane L<16 -> K={0,1}, lane L>=16 -> K={2,3}, N = L&15.
// ---------------------------------------------------------------------------
__global__ void armature_build_transforms(const float* __restrict__ quat,
                                          const float* __restrict__ scales,
                                          const float* __restrict__ offset,
                                          const float* __restrict__ relt,
                                          const int*   __restrict__ parents,
                                          float* __restrict__ outT,
                                          float* __restrict__ outScales,
                                          float* __restrict__ Btab) {
    __shared__ float Ts[NUM_BONES][16];
    const int tid = threadIdx.x;

    if (tid == 0) {
        for (int i = 0; i < NUM_BONES; ++i) {
            float qw = quat[i * 4 + 0], qx = quat[i * 4 + 1];
            float qy = quat[i * 4 + 2], qz = quat[i * 4 + 3];
            float inv = rsqrtf(qw * qw + qx * qx + qy * qy + qz * qz);
            float w = qw * inv, x = qx * inv, y = qy * inv, z = qz * inv;
            // Rotation matrix R (as in reference), then M rows = R^T rows.
            float r00 = 1.f - 2.f * (y * y + z * z);
            float r01 = 2.f * (x * y - w * z);
            float r02 = 2.f * (x * z + w * y);
            float r10 = 2.f * (x * y + w * z);
            float r11 = 1.f - 2.f * (x * x + z * z);
            float r12 = 2.f * (y * z - w * x);
            float r20 = 2.f * (x * z - w * y);
            float r21 = 2.f * (y * z + w * x);
            float r22 = 1.f - 2.f * (x * x + y * y);
            float* M = &Ts[i][0];
            M[0] = r00; M[1] = r10; M[2]  = r20; M[3]  = 0.f;
            M[4] = r01; M[5] = r11; M[6]  = r21; M[7]  = 0.f;
            M[8] = r02; M[9] = r12; M[10] = r22; M[11] = 0.f;
            int p = parents[i];
            float t0, t1, t2, t3;
            if (p < 0) {
                t0 = relt[i * 4 + 0] + offset[0];
                t1 = relt[i * 4 + 1] + offset[1];
                t2 = relt[i * 4 + 2] + offset[2];
                t3 = 1.f;
            } else {
                // NOTE: bug preserved — parent transform is the UNSCALED one.
                const float* Mp = &Ts[p][0];
                float a0 = relt[i * 4 + 0], a1 = relt[i * 4 + 1];
                float a2 = relt[i * 4 + 2], a3 = relt[i * 4 + 3];
                t0 = a0 * Mp[0] + a1 * Mp[4] + a2 * Mp[8]  + a3 * Mp[12];
                t1 = a0 * Mp[1] + a1 * Mp[5] + a2 * Mp[9]  + a3 * Mp[13];
                t2 = a0 * Mp[2] + a1 * Mp[6] + a2 * Mp[10] + a3 * Mp[14];
                t3 = a0 * Mp[3] + a1 * Mp[7] + a2 * Mp[11] + a3 * Mp[15];
            }
            M[12] = t0; M[13] = t1; M[14] = t2; M[15] = t3;
        }
    }
    __syncthreads();

    // T passthrough to output
    for (int idx = tid; idx < NUM_BONES * 16; idx += blockDim.x)
        outT[idx] = (&Ts[0][0])[idx];
    // pose_bone_scales passthrough (output #3)
    for (int idx = tid; idx < 32 * 3; idx += blockDim.x)
        outScales[idx] = scales[idx];
    // Pre-swizzled B operand table
    for (int b = tid; b < NUM_BONES; b += blockDim.x) {
        for (int L = 0; L < 32; ++L) {
            int col = L & 15;
            int k0  = (L >> 4) * 2;
            float v0 = (col < 4) ? Ts[b][(k0 + 0) * 4 + col] : 0.f;
            float v1 = (col < 4) ? Ts[b][(k0 + 1) * 4 + col] : 0.f;
            Btab[b * 64 + 2 * L + 0] = v0;
            Btab[b * 64 + 2 * L + 1] = v1;
        }
    }
}

// ---------------------------------------------------------------------------
// Kernel 2: LBS reduce, (nverts x 256) @ (256 x 16[pad]) via
// V_WMMA_F32_16X16X4_F32. One wave32 = one 16-vertex tile, K-loop = 64 bones.
// Global loads use uniform-SGPR-base + 32-bit lane offset (saddr form) so the
// unrolled loop issues loads with no per-load 64-bit address VALU / xcnt wait.
// Streaming vbi loads are non-temporal (410 MB once); B table stays in LDS/L2.
// ---------------------------------------------------------------------------
__global__ void __launch_bounds__(256)
armature_lbs_wmma(const float* __restrict__ vbi,
                  const float* __restrict__ Btab,
                  float* __restrict__ outv,
                  int nverts) {
    __shared__ float sB[NUM_BONES * 64];  // 16 KB
    {
        const v4f* src = (const v4f*)Btab;
        v4f*       dst = (v4f*)sB;
        for (int i = threadIdx.x; i < NUM_BONES * 64 / 4; i += blockDim.x)
            dst[i] = src[i];
    }
    __syncthreads();

    const int lane = threadIdx.x & 31;
    const long tile  = (long)blockIdx.x * 8 + (threadIdx.x >> 5);
    const long vbase = tile * 16;
    if (vbase >= nverts) return;  // wave-uniform: EXEC stays all-1s for WMMA

    const int vrow  = lane & 15;   // A-matrix M row for this lane
    const int khalf = lane >> 4;   // 0 -> K=0,1  ; 1 -> K=2,3
    long v = vbase + vrow;
    if (v >= nverts) v = nverts - 1;  // clamp (store-guarded below)

    // lane-varying 32-bit byte offset within one bone slab (max ~6.4 MB)
    const unsigned voff = (unsigned)v * 16u + (unsigned)khalf * 8u;
    const size_t   bstride = (size_t)nverts * 16u;   // bytes between bone slabs
    const char*    pb = (const char*)vbi;            // uniform base
    const float*   bptr = &sB[lane * 2];

    v8f acc = {0.f, 0.f, 0.f, 0.f, 0.f, 0.f, 0.f, 0.f};
#pragma unroll 8
    for (int b = 0; b < NUM_BONES; ++b) {
        // uniform 64-bit base (SGPR) + divergent 32-bit offset -> saddr form
        v2f A = __builtin_nontemporal_load(
            (const v2f*)(pb + (size_t)b * bstride + voff));
        v2f B = *(const v2f*)(bptr + b * 64);  // conflict-free ds_load_b64
        acc = __builtin_amdgcn_wmma_f32_16x16x4_f32(
            /*neg_a=*/false, A, /*neg_b=*/false, B,
            /*c_mod=*/(short)0, acc, /*reuse_a=*/false, /*reuse_b=*/false);
    }

    // D layout: VGPR r -> (lanes 0-15: M=r, lanes 16-31: M=8+r), N = lane&15.
    const int col = lane & 15;
    if (col >= 3) return;  // after all WMMAs: EXEC change is safe now

    const long rbase = vbase + (long)khalf * 8;
    if (vbase + 16 <= (long)nverts) {
        // full tile: 8 stores at immediate offsets from one base pointer
        float* o = outv + (size_t)rbase * 3 + col;
        o[0]  = acc[0]; o[3]  = acc[1]; o[6]  = acc[2]; o[9]  = acc[3];
        o[12] = acc[4]; o[15] = acc[5]; o[18] = acc[6]; o[21] = acc[7];
    } else {
#pragma unroll
        for (int r = 0; r < 8; ++r) {
            long vr = rbase + r;
            if (vr < nverts) outv[vr * 3 + col] = acc[r];
        }
    }
}

// ---------------------------------------------------------------------------
extern "C" void kernel_launch(void* const* d_in, const int* in_sizes, int n_in,
                              void* d_out, int out_size, void* d_ws, size_t ws_size,
                              hipStream_t stream) {
    const float* quat    = (const float*)d_in[4];   // pose_bone_quaternions [64,4]
    const float* scales  = (const float*)d_in[5];   // pose_bone_scales [32,3]
    const float* offset  = (const float*)d_in[6];   // offset [3]
    const float* relt    = (const float*)d_in[7];   // relative translations [64,4]
    const float* vbi     = (const float*)d_in[9];   // vertices_bone_inverted [64,V,4]
    const int*   parents = (const int*)d_in[10];    // bone_parents [64]

    const int nverts = in_sizes[9] / (NUM_BONES * 4);

    float* outv = (float*)d_out;                         // verts [V,3]
    float* outT = outv + (size_t)nverts * 3;             // T [64,4,4]
    float* outS = outT + NUM_BONES * 16;                 // scales [32,3]
    float* Btab = (float*)d_ws;                          // 64*64 floats (16 KB)

    hipLaunchKernelGGL(armature_build_transforms, dim3(1), dim3(64), 0, stream,
                       quat, scales, offset, relt, parents, outT, outS, Btab);

    const long tiles  = ((long)nverts + 15) / 16;
    const int  blocks = (int)((tiles + 7) / 8);
    hipLaunchKernelGGL(armature_lbs_wmma, dim3(blocks), dim3(256), 0, stream,
                       vbi, Btab, outv, nverts);
}